// SpectralLayerNorm_76785425318190
// MI455X (gfx1250) — compile-verified
//
#include <hip/hip_runtime.h>

#define LN_EPS 1e-5f
#define D_DIM  2048
#define TOK_F4 ((D_DIM * 2) / 4)   // 1024 float4 per token = 16 KB
#define BLOCK  256
#define CHUNKS (TOK_F4 / BLOCK)    // 4 float4 per thread

typedef __attribute__((ext_vector_type(4))) float f4;
typedef __attribute__((ext_vector_type(2))) float f2;
typedef __attribute__((ext_vector_type(2))) float v2f;
typedef __attribute__((ext_vector_type(8))) float v8f;

#if __has_builtin(__builtin_amdgcn_global_load_async_to_lds_b128)
#define USE_ASYNC_LDS 1
// Builtin expects: (v4i addrspace(1)*, v4i addrspace(3)*, imm offset, imm cpol)
typedef __attribute__((__vector_size__(4 * sizeof(int)))) int v4i;
typedef __attribute__((address_space(1))) v4i* as1_v4i;
typedef __attribute__((address_space(3))) v4i* as3_v4i;
#endif

#if __has_builtin(__builtin_amdgcn_wmma_f32_16x16x4_f32)
#define USE_WMMA_RED 1
#endif

__launch_bounds__(BLOCK)
__global__ void spectral_ln_kernel(const float* __restrict__ x,
                                   const float* __restrict__ gamma,
                                   const float* __restrict__ beta,
                                   float* __restrict__ out)
{
    const int tid  = threadIdx.x;
    const int lane = tid & 31;
    const int wave = tid >> 5;
    const size_t tok = blockIdx.x;

    const f4* __restrict__ xin = reinterpret_cast<const f4*>(x) + tok * TOK_F4;
    f4* __restrict__ yout      = reinterpret_cast<f4*>(out)     + tok * TOK_F4;

    f4 v[CHUNKS];

#if USE_ASYNC_LDS
    // Stage this token's 16KB into LDS via the CDNA5 async global->LDS path.
    __shared__ f4 sbuf[TOK_F4];
    #pragma unroll
    for (int k = 0; k < CHUNKS; ++k) {
        const int idx = tid + k * BLOCK;
        __builtin_amdgcn_global_load_async_to_lds_b128(
            (as1_v4i)(size_t)(xin + idx),
            (as3_v4i)(unsigned int)(size_t)(sbuf + idx),
            /*imm offset*/0, /*cpol*/0);
    }
    asm volatile("s_wait_asynccnt 0" ::: "memory");
    // Each lane reads back exactly the chunks it issued: no barrier required.
    #pragma unroll
    for (int k = 0; k < CHUNKS; ++k) v[k] = sbuf[tid + k * BLOCK];
#else
    #pragma unroll
    for (int k = 0; k < CHUNKS; ++k)
        v[k] = __builtin_nontemporal_load(&xin[tid + k * BLOCK]);
#endif

    // Raw moments over this thread's 8 complex elements.
    float sr = 0.f, si = 0.f, srr = 0.f, sri = 0.f, sii = 0.f;
    #pragma unroll
    for (int k = 0; k < CHUNKS; ++k) {
        const float r0 = v[k].x, i0 = v[k].y, r1 = v[k].z, i1 = v[k].w;
        sr  += r0 + r1;
        si  += i0 + i1;
        srr += r0 * r0 + r1 * r1;
        sri += r0 * i0 + r1 * i1;
        sii += i0 * i0 + i1 * i1;
    }

    // wave32 butterfly reduction.
    #pragma unroll
    for (int off = 16; off >= 1; off >>= 1) {
        sr  += __shfl_xor(sr,  off, 32);
        si  += __shfl_xor(si,  off, 32);
        srr += __shfl_xor(srr, off, 32);
        sri += __shfl_xor(sri, off, 32);
        sii += __shfl_xor(sii, off, 32);
    }

    // Per-wave partials -> LDS, zero-padded so WMMA A rows 5..15 are clean.
    __shared__ float red[8][16];
    if (tid < 128) red[tid >> 4][tid & 15] = 0.0f;
    __syncthreads();
    if (lane == 0) {
        red[wave][0] = sr;  red[wave][1] = si;  red[wave][2] = srr;
        red[wave][3] = sri; red[wave][4] = sii;
    }
    __syncthreads();

    float Sr, Si, Srr, Sri, Sii;
#if USE_WMMA_RED
    {
        // Cross-wave sum as D = A*ones + C with V_WMMA_F32_16X16X4_F32.
        // A layout (16x4 f32): A[m,k] -> lane m+16*(k>=2), vgpr (k&1).
        // Row m = quantity m, K = wave index. Two WMMAs cover waves 0..3, 4..7.
        const int m  = tid & 15;
        const int hi = (lane >> 4) & 1;
        v2f a0, a1, ones;
        a0.x = red[hi ? 2 : 0][m];
        a0.y = red[hi ? 3 : 1][m];
        a1.x = red[hi ? 6 : 4][m];
        a1.y = red[hi ? 7 : 5][m];
        ones.x = 1.0f; ones.y = 1.0f;
        v8f c = {0.f, 0.f, 0.f, 0.f, 0.f, 0.f, 0.f, 0.f};
        c = __builtin_amdgcn_wmma_f32_16x16x4_f32(false, a0, false, ones,
                                                  (short)0, c, false, false);
        c = __builtin_amdgcn_wmma_f32_16x16x4_f32(false, a1, false, ones,
                                                  (short)0, c, false, false);
        // D row q lives in vgpr q for lanes 0..15 (replicated over N).
        Sr  = __shfl(c[0], 0, 32);
        Si  = __shfl(c[1], 0, 32);
        Srr = __shfl(c[2], 0, 32);
        Sri = __shfl(c[3], 0, 32);
        Sii = __shfl(c[4], 0, 32);
    }
#else
    Sr = Si = Srr = Sri = Sii = 0.f;
    #pragma unroll
    for (int w = 0; w < 8; ++w) {
        Sr  += red[w][0]; Si  += red[w][1]; Srr += red[w][2];
        Sri += red[w][3]; Sii += red[w][4];
    }
#endif

    // Centered covariance from raw moments; analytic SPD 2x2 inverse sqrt.
    const float invD   = 1.0f / (float)D_DIM;
    const float invDm1 = 1.0f / (float)(D_DIM - 1);
    const float mr = Sr * invD;
    const float mi = Si * invD;
    const float va = (Srr - (float)D_DIM * mr * mr) * invDm1 + LN_EPS;
    const float vc = (Sii - (float)D_DIM * mi * mi) * invDm1 + LN_EPS;
    const float vb = (Sri - (float)D_DIM * mr * mi) * invDm1;

    const float s   = __builtin_sqrtf(fmaxf(va * vc - vb * vb, 1e-30f));
    const float t   = __builtin_sqrtf(va + vc + 2.0f * s);
    const float inv = 1.0f / (s * t);
    const float w00 = (vc + s) * inv;
    const float w01 = -vb * inv;
    const float w11 = (va + s) * inv;

    const f4* __restrict__ beta4 = reinterpret_cast<const f4*>(beta);

    #pragma unroll
    for (int k = 0; k < CHUNKS; ++k) {
        const int idx = tid + k * BLOCK;          // complex channels 2*idx, 2*idx+1
        const f2* __restrict__ g2p = reinterpret_cast<const f2*>(gamma + 6 * (size_t)idx);
        const f2 g01 = g2p[0], g23 = g2p[1], g45 = g2p[2];
        const f4 bt = beta4[idx];

        f4 o;
        {
            const float xr = v[k].x - mr, xi = v[k].y - mi;
            const float wr = w00 * xr + w01 * xi;
            const float wi = w01 * xr + w11 * xi;
            o.x = g01.x * wr + g01.y * wi + bt.x;   // g0a, g1a
            o.y = g01.y * wr + g23.x * wi + bt.y;   // g1a, g2a
        }
        {
            const float xr = v[k].z - mr, xi = v[k].w - mi;
            const float wr = w00 * xr + w01 * xi;
            const float wi = w01 * xr + w11 * xi;
            o.z = g23.y * wr + g45.x * wi + bt.z;   // g0b, g1b
            o.w = g45.x * wr + g45.y * wi + bt.w;   // g1b, g2b
        }
        __builtin_nontemporal_store(o, &yout[idx]);
    }
}

extern "C" void kernel_launch(void* const* d_in, const int* in_sizes, int n_in,
                              void* d_out, int out_size, void* d_ws, size_t ws_size,
                              hipStream_t stream) {
    (void)n_in; (void)out_size; (void)d_ws; (void)ws_size;
    const float* x     = (const float*)d_in[0];
    const float* gamma = (const float*)d_in[1];
    const float* beta  = (const float*)d_in[2];
    float* out         = (float*)d_out;
    const int tokens = in_sizes[0] / (D_DIM * 2);   // B*S = 16384
    spectral_ln_kernel<<<dim3(tokens), dim3(BLOCK), 0, stream>>>(x, gamma, beta, out);
}